// NN_71451075936307
// MI455X (gfx1250) — compile-verified
//
#include <hip/hip_runtime.h>
#include <math.h>

typedef __attribute__((ext_vector_type(2))) float    v2f;
typedef __attribute__((ext_vector_type(8))) float    v8f;
typedef __attribute__((ext_vector_type(4))) unsigned u32x4;
typedef __attribute__((ext_vector_type(8))) int      i32x8;
typedef __attribute__((ext_vector_type(4))) int      i32x4;

#define BB 4096
#define SS 64
#define HH 64
#define NSEL 4
#define PP 64
// out layout: proto_st [BB*NSEL*HH] | residuals [BB*NSEL*HH] | loss [1] | indexes [BB*NSEL]
#define OUT_PROTO 0
#define OUT_RES   (BB*NSEL*HH)
#define OUT_LOSS  (2*BB*NSEL*HH)
#define OUT_IDX   (2*BB*NSEL*HH + 1)

// ---------------------------------------------------------------- utilities

__device__ __forceinline__ unsigned rotl32(unsigned x, unsigned r) {
    return (x << r) | (x >> (32u - r));
}

// threefry2x32-20, key = (0,1) (jax.random.key(1)), ctr = (0, idx)
__device__ __forceinline__ float gumbel_noise(unsigned idx) {
    unsigned ks0 = 0u, ks1 = 1u, ks2 = 0x1BD11BDAu ^ ks0 ^ ks1;
    unsigned x0 = 0u + ks0, x1 = idx + ks1;
#define TF_ROUND(r) { x0 += x1; x1 = rotl32(x1, (r)); x1 ^= x0; }
    TF_ROUND(13) TF_ROUND(15) TF_ROUND(26) TF_ROUND(6)
    x0 += ks1; x1 += ks2 + 1u;
    TF_ROUND(17) TF_ROUND(29) TF_ROUND(16) TF_ROUND(24)
    x0 += ks2; x1 += ks0 + 2u;
    TF_ROUND(13) TF_ROUND(15) TF_ROUND(26) TF_ROUND(6)
    x0 += ks0; x1 += ks1 + 3u;
    TF_ROUND(17) TF_ROUND(29) TF_ROUND(16) TF_ROUND(24)
    x0 += ks1; x1 += ks2 + 4u;
    TF_ROUND(13) TF_ROUND(15) TF_ROUND(26) TF_ROUND(6)
    x0 += ks2; x1 += ks0 + 5u;
#undef TF_ROUND
    float u = (float)(x0 >> 8) * 5.9604645e-08f;      // [0,1) at 2^-24
    u = fmaxf(u, 1.1754944e-38f);
    return -__logf(fmaxf(-__logf(u), 1.1754944e-38f));
}

// LDS byte offset of a __shared__ object (generic -> AS3 addrspacecast)
__device__ __forceinline__ unsigned lds_off(const void* p) {
    return (unsigned)(unsigned long long)
           (const __attribute__((address_space(3))) void*)p;
}

// Issue one 1-D TDM load: nelems f32 from global gptr to the LDS byte offset lds_byte_off.
// D# per CDNA5 ISA 8.3/8.4: group0 {count=1, lds_addr, global_addr, type=2},
// group1 {data_size=4B, tensor_dim0=tile_dim0=nelems, dim1=1, stride0=nelems}.
__device__ __forceinline__ void tdm_load_f32_1d(unsigned lds_byte_off,
                                                const void* gptr,
                                                unsigned nelems) {
    unsigned long long ga = (unsigned long long)gptr;
    u32x4 g0;
    g0[0] = 1u;                                           // count=1, user mode
    g0[1] = lds_byte_off;                                 // lds_addr
    g0[2] = (unsigned)ga;                                 // global_addr[31:0]
    g0[3] = (unsigned)((ga >> 32) & 0x1FFFFFFu) | (2u << 30); // [56:32] | type=2
    i32x8 g1;
    g1[0] = (int)(2u << 16);                              // data_size = 4 bytes
    g1[1] = (int)((nelems & 0xFFFFu) << 16);              // tensor_dim0[15:0]
    g1[2] = (int)(((nelems >> 16) & 0xFFFFu) | (1u << 16)); // tensor_dim0[31:16], tensor_dim1=1
    g1[3] = (int)((nelems & 0xFFFFu) << 16);              // tile_dim0 = nelems (<=65535)
    g1[4] = 0;                                            // tile_dim1=0 (unused), tile_dim2=0
    g1[5] = (int)nelems;                                  // tensor_dim0_stride[31:0]
    g1[6] = 0;                                            // stride0[47:32], stride1[15:0]
    g1[7] = 0;
    i32x4 z4 = {0, 0, 0, 0};
    i32x8 z8 = {0, 0, 0, 0, 0, 0, 0, 0};
    __builtin_amdgcn_tensor_load_to_lds(g0, g1, z4, z4, z8, 0);
}

// ---------------------------------------------------------------- kernel 0: zero loss slot

__global__ void zero_loss_kernel(float* p) {
    if (threadIdx.x == 0 && blockIdx.x == 0) p[0] = 0.0f;
}

// ---------------------------------------------------------------- kernel 1: positional encoding -> ws

__global__ void pe_kernel(float* __restrict__ PE) {
    int t = blockIdx.x * blockDim.x + threadIdx.x;     // 0..4095
    int s = t >> 6, h = t & 63;
    int he = h & ~1;
    float div = __expf((float)he * (-9.210340371976184f / 64.0f)); // -ln(10000)/H
    float a = (float)s * div;
    PE[t] = (h & 1) ? __cosf(a) : __sinf(a);
}

// ---------------------------------------------------------------- kernel 2: M_n = Wq_n^T @ Wk_n  (f32 WMMA)

__global__ __launch_bounds__(32)
void precompute_M(const float* __restrict__ Wq, const float* __restrict__ Wk,
                  float* __restrict__ M) {
    const int n  = blockIdx.x;
    const int i0 = blockIdx.y * 16;
    const int j0 = blockIdx.z * 16;
    const int l  = threadIdx.x;
    const int hf = l >> 4, lo = l & 15;

    const float* wq = Wq + n * 128 * HH;
    const float* wk = Wk + n * 128 * HH;

    v8f c = {};
    #pragma unroll 4
    for (int k0 = 0; k0 < 128; k0 += 4) {
        const int ka = k0 + 2 * hf;
        v2f a, b;
        a.x = wq[(ka    ) * HH + i0 + lo];   // A-frag: lane lo = row, VGPR0=K{0,2}, VGPR1=K{1,3}
        a.y = wq[(ka + 1) * HH + i0 + lo];
        b.x = wk[(ka    ) * HH + j0 + lo];   // B-frag: lane lo = col, same K striping
        b.y = wk[(ka + 1) * HH + j0 + lo];
        c = __builtin_amdgcn_wmma_f32_16x16x4_f32(false, a, false, b,
                                                  (short)0, c, false, false);
    }
    #pragma unroll
    for (int v = 0; v < 8; v++) {
        int mi = i0 + v + 8 * hf;            // C: lane lo = N, vgpr v -> M = v + 8*hf
        M[n * (HH * HH) + mi * HH + (j0 + lo)] = c[v];
    }
}

// ---------------------------------------------------------------- kernel 3: fused per-batch pipeline
// one block (128 threads = 4 waves) per batch b. LDS filled by the Tensor Data Mover.

__global__ __launch_bounds__(128)
void fused_select_vq(const float* __restrict__ x,
                     const float* __restrict__ prototypes,
                     const float* __restrict__ Mws,   // [4][64][64]
                     const float* __restrict__ PEws,  // [64][64]
                     float* __restrict__ out) {
    __shared__ float sX[SS * HH];        // raw x tile (TDM)
    __shared__ float sPE[SS * HH];       // positional encoding (TDM)
    __shared__ float sProto[PP * HH];    // prototypes (TDM)
    __shared__ float sV[NSEL * HH];
    __shared__ float sLog[NSEL * SS];
    __shared__ float sSum[HH];
    __shared__ float sPart[2 * HH];
    __shared__ float sXsel[NSEL * HH];
    __shared__ int   sIdx[NSEL];
    __shared__ int   sPidx[NSEL];
    __shared__ float sRed[128];

    const int b    = blockIdx.x;
    const int t    = threadIdx.x;
    const int lane = t & 31;
    const int wave = t >> 5;             // 0..3 == head n
    const int hf   = lane >> 4, lo = lane & 15;

    // ---- 1. TDM fill of LDS: x tile, PE, prototypes (wave 0 only; scalar branch so
    //         waves 1-3 jump over it -- TDM ignores EXEC, masking is not enough)
    if (__builtin_amdgcn_readfirstlane(t) < 32) {
        tdm_load_f32_1d(lds_off(sX),     x + (size_t)b * (SS * HH), SS * HH);
        tdm_load_f32_1d(lds_off(sPE),    PEws,                      SS * HH);
        tdm_load_f32_1d(lds_off(sProto), prototypes,                PP * HH);
        __builtin_amdgcn_s_wait_tensorcnt(0);
    }
    __syncthreads();

    // ---- 2. possum[h] = sum_s (x+PE)[s][h]
    {
        int h = t & 63, g = t >> 6;
        float acc = 0.0f;
        #pragma unroll 8
        for (int s = 32 * g; s < 32 * g + 32; s++)
            acc += sX[s * HH + h] + sPE[s * HH + h];
        sPart[g * HH + h] = acc;
    }
    __syncthreads();
    if (t < HH) sSum[t] = sPart[t] + sPart[HH + t];
    __syncthreads();

    // ---- 3. v[n][i] = (M_n @ possum)[i] / 512   (1/(S*sqrt(H)))
    #pragma unroll
    for (int r = 0; r < 2; r++) {
        int o = t + 128 * r;
        int n = o >> 6, i = o & 63;
        const float* mr = Mws + n * (HH * HH) + i * HH;
        float acc = 0.0f;
        #pragma unroll 8
        for (int k = 0; k < HH; k++) acc += mr[k] * sSum[k];
        sV[o] = acc * (1.0f / 512.0f);
    }
    __syncthreads();

    // ---- 4. logits[n][s] = pos[s]·v[n]  via f32 WMMA (wave w -> s rows 16w..16w+15)
    {
        const int s0 = 16 * wave;
        v8f c = {};
        #pragma unroll 4
        for (int k0 = 0; k0 < HH; k0 += 4) {
            const int ka  = k0 + 2 * hf;
            const int row = (s0 + lo) * HH;
            v2f a, bf;
            a.x  = sX[row + ka]     + sPE[row + ka];       // pos on the fly
            a.y  = sX[row + ka + 1] + sPE[row + ka + 1];
            bf.x = (lo < NSEL) ? sV[lo * HH + ka]     : 0.0f;  // B[k][n]=v[n][k], N pad 16
            bf.y = (lo < NSEL) ? sV[lo * HH + ka + 1] : 0.0f;
            c = __builtin_amdgcn_wmma_f32_16x16x4_f32(false, a, false, bf,
                                                      (short)0, c, false, false);
        }
        if (lo < NSEL) {
            #pragma unroll
            for (int vv = 0; vv < 8; vv++) {
                int srow = s0 + vv + 8 * hf;
                sLog[lo * SS + srow] = c[vv];
            }
        }
    }
    __syncthreads();

    // ---- 5. argmax_s(logits + gumbel)  (one wave per head, shuffle reduce)
    {
        const int n = wave;
        float best = -INFINITY; int bidx = 0;
        #pragma unroll
        for (int rep = 0; rep < 2; rep++) {
            int s = lane + 32 * rep;
            float v = sLog[n * SS + s] +
                      gumbel_noise(((unsigned)b * NSEL + (unsigned)n) * SS + (unsigned)s);
            if (v > best || (v == best && s < bidx)) { best = v; bidx = s; }
        }
        #pragma unroll
        for (int m = 16; m > 0; m >>= 1) {
            float ov = __shfl_xor(best, m, 32);
            int   oi = __shfl_xor(bidx, m, 32);
            if (ov > best || (ov == best && oi < bidx)) { best = ov; bidx = oi; }
        }
        if (lane == 0) sIdx[n] = bidx;
    }
    __syncthreads();

    // ---- 6. x_sel[n][h] = x[b, idx[n], h]  (raw x -- alphas is exactly one-hot fwd)
    #pragma unroll
    for (int r = 0; r < 2; r++) {
        int o = t + 128 * r;
        int n = o >> 6, h = o & 63;
        sXsel[o] = sX[sIdx[n] * HH + h];
    }
    __syncthreads();

    // ---- 7. VQ: argmin_p ( 0.5*|proto_p|^2 - x_sel·proto_p )  (one wave per head)
    {
        const int n = wave;
        float bestd = INFINITY; int bp = 0;
        #pragma unroll
        for (int rep = 0; rep < 2; rep++) {
            int p = lane + 32 * rep;
            float acc = 0.0f;
            #pragma unroll 8
            for (int h = 0; h < HH; h++) {
                float pv = sProto[p * HH + h];
                acc += pv * (0.5f * pv - sXsel[n * HH + h]);
            }
            if (acc < bestd || (acc == bestd && p < bp)) { bestd = acc; bp = p; }
        }
        #pragma unroll
        for (int m = 16; m > 0; m >>= 1) {
            float od = __shfl_xor(bestd, m, 32);
            int   op = __shfl_xor(bp, m, 32);
            if (od < bestd || (od == bestd && op < bp)) { bestd = od; bp = op; }
        }
        if (lane == 0) sPidx[n] = bp;
    }
    __syncthreads();

    // ---- 8. outputs: proto_st = proto, residual = x_sel - proto, loss += 1.25*mean(res^2)
    float lacc = 0.0f;
    #pragma unroll
    for (int r = 0; r < 2; r++) {
        int o = t + 128 * r;
        int n = o >> 6, h = o & 63;
        float pv  = sProto[sPidx[n] * HH + h];
        float xs  = sXsel[o];
        float res = xs - pv;
        out[OUT_PROTO + (size_t)b * (NSEL * HH) + o] = pv;
        out[OUT_RES   + (size_t)b * (NSEL * HH) + o] = res;
        lacc += res * res;
    }
    sRed[t] = lacc;
    __syncthreads();
    #pragma unroll
    for (int stp = 64; stp > 0; stp >>= 1) {
        if (t < stp) sRed[t] += sRed[t + stp];
        __syncthreads();
    }
    if (t == 0)
        atomicAdd(out + OUT_LOSS, sRed[0] * (1.25f / (float)(BB * NSEL * HH)));
    if (t < NSEL)
        out[OUT_IDX + (size_t)b * NSEL + t] = (float)sIdx[t];
}

// ---------------------------------------------------------------- launch

extern "C" void kernel_launch(void* const* d_in, const int* in_sizes, int n_in,
                              void* d_out, int out_size, void* d_ws, size_t ws_size,
                              hipStream_t stream) {
    (void)in_sizes; (void)n_in; (void)out_size; (void)ws_size;
    const float* x          = (const float*)d_in[0];
    const float* Wq         = (const float*)d_in[1];
    const float* Wk         = (const float*)d_in[2];
    const float* prototypes = (const float*)d_in[3];
    float* out = (float*)d_out;

    float* M  = (float*)d_ws;        // 4*64*64 floats
    float* PE = M + NSEL * HH * HH;  // 64*64 floats

    zero_loss_kernel<<<1, 1, 0, stream>>>(out + OUT_LOSS);
    pe_kernel<<<16, 256, 0, stream>>>(PE);
    dim3 gM(NSEL, 4, 4);
    precompute_M<<<gM, 32, 0, stream>>>(Wq, Wk, M);
    fused_select_vq<<<BB, 128, 0, stream>>>(x, prototypes, M, PE, out);
}